// Flow_25108378812712
// MI455X (gfx1250) — compile-verified
//
#include <hip/hip_runtime.h>
#include <hip/hip_bf16.h>

// ---------------- problem constants ----------------
#define DSZ 128
#define DHW (128*128*128)
#define CIN 16
#define NDOWN 64
#define N3 (64*64*64)

typedef __attribute__((ext_vector_type(2))) float v2f;
typedef __attribute__((ext_vector_type(8))) float v8f;

#if __has_builtin(__builtin_amdgcn_global_load_async_to_lds_b32)
#define HAVE_ASYNC_LDS 1
#else
#define HAVE_ASYNC_LDS 0
#endif

__device__ __forceinline__ void wait_asynccnt0() {
#if __has_builtin(__builtin_amdgcn_s_wait_asynccnt)
    __builtin_amdgcn_s_wait_asynccnt(0);
#else
    asm volatile("s_wait_asynccnt 0x0" ::: "memory");
#endif
}

// ---------------- conv3d 16->3, k=3, pad=1, via WMMA f32 16x16x4 ----------------
// Tile: 16(x) x 8(y) x 2(z) outputs per block; halo 18x10x4.
#define TX 16
#define TY 8
#define TZ 2
#define HX 18
#define HY 10
#define HZ 4
#define CPAD 17   // pad channel dim to 17 dwords: stride 17 is coprime with 64 banks

__global__ __launch_bounds__(256, 1)
void conv3d_wmma_kernel(const float* __restrict__ feat,
                        const float* __restrict__ wgt,
                        const float* __restrict__ bias,
                        float* __restrict__ out /* pos_flow [3][128^3] */) {
    __shared__ float sF[HZ][HY][HX][CPAD];
    __shared__ float sW[4][27 * 16];     // row 3 = zeros (for WMMA columns N>=3)

    const int tid = threadIdx.x;
    const int x0 = blockIdx.x * TX;
    const int y0 = blockIdx.y * TY;
    const int z0 = blockIdx.z * TZ;

    // ---- stage weights: sW[c][koff*16+ic] = W[c][ic][koff], row 3 zero ----
    for (int i = tid; i < 4 * 432; i += 256) {
        int c = i / 432, k = i - c * 432;
        int koff = k >> 4, ic = k & 15;
        sW[c][k] = (c < 3) ? wgt[(c * CIN + ic) * 27 + koff] : 0.0f;
    }

    // ---- stage features with halo (zero pad outside volume) ----
#if HAVE_ASYNC_LDS
    {
        typedef __attribute__((address_space(1))) int g_int;
        typedef __attribute__((address_space(3))) int l_int;
        g_int* gbase = (g_int*)feat;   // features viewed as global dwords
        const bool interior = (x0 > 0) && (x0 + TX < DSZ) &&
                              (y0 > 0) && (y0 + TY < DSZ) &&
                              (z0 > 0) && (z0 + TZ < DSZ);
        if (interior) {
            // all halo voxels in range: unpredicated async global->LDS copies
            for (int ic = 0; ic < CIN; ++ic) {
                for (int i = tid; i < HZ * HY * HX; i += 256) {
                    int lx = i % HX;
                    int ly = (i / HX) % HY;
                    int lz = i / (HX * HY);
                    int gx = x0 + lx - 1, gy = y0 + ly - 1, gz = z0 + lz - 1;
                    int goff = ((ic * DSZ + gz) * DSZ + gy) * DSZ + gx;  // dword index
                    l_int* lp = (l_int*)&sF[lz][ly][lx][ic];
                    __builtin_amdgcn_global_load_async_to_lds_b32(gbase + goff, lp, 0, 0);
                }
            }
        } else {
            // zero-fill LDS tile, then predicated async copies (EXEC-masked lanes skip)
            float* sflat = &sF[0][0][0][0];
            for (int i = tid; i < HZ * HY * HX * CPAD; i += 256)
                sflat[i] = 0.0f;
            __syncthreads();   // block-uniform branch: legal; orders zero-fill vs async writes
            for (int ic = 0; ic < CIN; ++ic) {
                for (int i = tid; i < HZ * HY * HX; i += 256) {
                    int lx = i % HX;
                    int ly = (i / HX) % HY;
                    int lz = i / (HX * HY);
                    int gx = x0 + lx - 1, gy = y0 + ly - 1, gz = z0 + lz - 1;
                    if ((unsigned)gx < 128u && (unsigned)gy < 128u && (unsigned)gz < 128u) {
                        int goff = ((ic * DSZ + gz) * DSZ + gy) * DSZ + gx;
                        l_int* lp = (l_int*)&sF[lz][ly][lx][ic];
                        __builtin_amdgcn_global_load_async_to_lds_b32(gbase + goff, lp, 0, 0);
                    }
                }
            }
        }
        wait_asynccnt0();      // all of this wave's async LDS writes landed
    }
#else
    for (int ic = 0; ic < CIN; ++ic) {
        for (int i = tid; i < HZ * HY * HX; i += 256) {
            int lx = i % HX;
            int ly = (i / HX) % HY;
            int lz = i / (HX * HY);
            int gx = x0 + lx - 1, gy = y0 + ly - 1, gz = z0 + lz - 1;
            float v = 0.0f;
            if ((unsigned)gx < 128u && (unsigned)gy < 128u && (unsigned)gz < 128u)
                v = feat[((ic * DSZ + gz) * DSZ + gy) * DSZ + gx];
            sF[lz][ly][lx][ic] = v;
        }
    }
#endif
    __syncthreads();

    const int lane = tid & 31;
    const int wave = tid >> 5;          // 0..7
    const int m    = lane & 15;         // M row (A) / N column (B, C/D)
    const int half = lane >> 4;         // K half-select
    const int nn   = (m < 3) ? m : 3;   // weight row; >=3 reads zero row

    // 16 (y,z) sub-tiles of 16 x-voxels each; 8 waves, 2 tiles per wave
    for (int t = wave; t < TY * TZ; t += 8) {
        const int ty = t % TY;
        const int tz = t / TY;
        v8f acc = {0.f, 0.f, 0.f, 0.f, 0.f, 0.f, 0.f, 0.f};
        for (int koff = 0; koff < 27; ++koff) {
            const int kd = koff / 9;
            const int kh = (koff / 3) % 3;
            const int kw = koff % 3;
            const float* fp = &sF[tz + kd][ty + kh][m + kw][0];
            const float* wp = &sW[nn][koff * 16];
#pragma unroll
            for (int ic0 = 0; ic0 < 16; ic0 += 4) {
                v2f a, b;
                a.x = fp[ic0 + 2 * half];
                a.y = fp[ic0 + 2 * half + 1];
                b.x = wp[ic0 + 2 * half];
                b.y = wp[ic0 + 2 * half + 1];
                // D(16x16) += A(16x4, voxels x ic) * B(4x16, ic x channel)
                acc = __builtin_amdgcn_wmma_f32_16x16x4_f32(
                    false, a, false, b, (short)0, acc, false, false);
            }
        }
        // lane m (<3) holds channel m; vgpr r holds voxel x0 + r + 8*half
        if (m < 3) {
            const float bz = bias[m];
            const int gz = z0 + tz, gy = y0 + ty;
            const int base = m * DHW + (gz * DSZ + gy) * DSZ + x0 + 8 * half;
#pragma unroll
            for (int r = 0; r < 8; ++r)
                out[base + r] = acc[r] + bz;
        }
    }
}

// ---------------- trilinear helpers ----------------
// In-range trilinear sample of one channel of a [dim^3] volume (coords in [0,dim-1])
__device__ __forceinline__ float tri_inrange(const float* __restrict__ v, int dim,
                                             float pz, float py, float px) {
    float fz = floorf(pz), fy = floorf(py), fx = floorf(px);
    int z0 = (int)fz, y0 = (int)fy, x0 = (int)fx;
    float wz = pz - fz, wy = py - fy, wx = px - fx;
    int z1 = min(z0 + 1, dim - 1), y1 = min(y0 + 1, dim - 1), x1 = min(x0 + 1, dim - 1);
    z0 = min(max(z0, 0), dim - 1); y0 = min(max(y0, 0), dim - 1); x0 = min(max(x0, 0), dim - 1);
    const float* p00 = v + (z0 * dim + y0) * dim;
    const float* p01 = v + (z0 * dim + y1) * dim;
    const float* p10 = v + (z1 * dim + y0) * dim;
    const float* p11 = v + (z1 * dim + y1) * dim;
    float c00 = p00[x0] * (1.f - wx) + p00[x1] * wx;
    float c01 = p01[x0] * (1.f - wx) + p01[x1] * wx;
    float c10 = p10[x0] * (1.f - wx) + p10[x1] * wx;
    float c11 = p11[x0] * (1.f - wx) + p11[x1] * wx;
    float c0 = c00 * (1.f - wy) + c01 * wy;
    float c1 = c10 * (1.f - wy) + c11 * wy;
    return c0 * (1.f - wz) + c1 * wz;
}

// ---------------- downsample pos_flow 128^3 -> 64^3, scale (1/2)*(1/128) ----------------
__global__ void downsample_kernel(const float* __restrict__ pf, float* __restrict__ f0) {
    int idx = blockIdx.x * blockDim.x + threadIdx.x;
    if (idx >= N3) return;
    int x = idx & 63, y = (idx >> 6) & 63, z = idx >> 12;
    const float r = 127.0f / 63.0f;
    float pz = z * r, py = y * r, px = x * r;
    const float s = 0.5f / 128.0f;
#pragma unroll
    for (int c = 0; c < 3; ++c)
        f0[c * N3 + idx] = tri_inrange(pf + c * DHW, DSZ, pz, py, px) * s;
}

// ---------------- one scaling-and-squaring step at 64^3: f = f + warp(f, f) ----------------
__global__ void vecint_step_kernel(const float* __restrict__ fin, float* __restrict__ fout) {
    int idx = blockIdx.x * blockDim.x + threadIdx.x;
    if (idx >= N3) return;
    int x = idx & 63, y = (idx >> 6) & 63, z = idx >> 12;
    float f0 = fin[idx], f1 = fin[N3 + idx], f2 = fin[2 * N3 + idx];
    float pz = z + f0, py = y + f1, px = x + f2;
    float fz = floorf(pz), fy = floorf(py), fx = floorf(px);
    int z0 = (int)fz, y0 = (int)fy, x0 = (int)fx;
    float wz = pz - fz, wy = py - fy, wx = px - fx;
    float s0 = 0.f, s1 = 0.f, s2 = 0.f;
#pragma unroll
    for (int cz = 0; cz < 2; ++cz)
#pragma unroll
        for (int cy = 0; cy < 2; ++cy)
#pragma unroll
            for (int cx = 0; cx < 2; ++cx) {
                int iz = z0 + cz, iy = y0 + cy, ix = x0 + cx;
                float w = (cz ? wz : 1.f - wz) * (cy ? wy : 1.f - wy) * (cx ? wx : 1.f - wx);
                if ((unsigned)iz < 64u && (unsigned)iy < 64u && (unsigned)ix < 64u) {
                    int o = (iz << 12) | (iy << 6) | ix;
                    s0 += w * fin[o];
                    s1 += w * fin[N3 + o];
                    s2 += w * fin[2 * N3 + o];
                }
            }
    fout[idx] = f0 + s0;
    fout[N3 + idx] = f1 + s1;
    fout[2 * N3 + idx] = f2 + s2;
}

// ---------------- upsample (x2, 64^3 -> 128^3) + warp label; write moved + flow ----------------
__global__ void finalize_kernel(const float* __restrict__ f, const float* __restrict__ label,
                                float* __restrict__ out /* [moved | flow] */) {
    int idx = blockIdx.x * blockDim.x + threadIdx.x;
    if (idx >= DHW) return;
    int x = idx & 127, y = (idx >> 7) & 127, z = idx >> 14;
    const float r = 63.0f / 127.0f;
    float sz = z * r, sy = y * r, sx = x * r;
    float fl0 = 2.0f * tri_inrange(f,            NDOWN, sz, sy, sx);
    float fl1 = 2.0f * tri_inrange(f + N3,       NDOWN, sz, sy, sx);
    float fl2 = 2.0f * tri_inrange(f + 2 * N3,   NDOWN, sz, sy, sx);
    out[DHW + 0 * DHW + idx] = fl0;
    out[DHW + 1 * DHW + idx] = fl1;
    out[DHW + 2 * DHW + idx] = fl2;

    // warp label with zero padding
    float pz = z + fl0, py = y + fl1, px = x + fl2;
    float fz = floorf(pz), fy = floorf(py), fx = floorf(px);
    int z0 = (int)fz, y0 = (int)fy, x0 = (int)fx;
    float wz = pz - fz, wy = py - fy, wx = px - fx;
    float acc = 0.f;
#pragma unroll
    for (int cz = 0; cz < 2; ++cz)
#pragma unroll
        for (int cy = 0; cy < 2; ++cy)
#pragma unroll
            for (int cx = 0; cx < 2; ++cx) {
                int iz = z0 + cz, iy = y0 + cy, ix = x0 + cx;
                float w = (cz ? wz : 1.f - wz) * (cy ? wy : 1.f - wy) * (cx ? wx : 1.f - wx);
                if ((unsigned)iz < 128u && (unsigned)iy < 128u && (unsigned)ix < 128u)
                    acc += w * label[((iz << 7) | iy) * 128 + ix];
            }
    out[idx] = acc;
}

// ---------------- launcher ----------------
extern "C" void kernel_launch(void* const* d_in, const int* in_sizes, int n_in,
                              void* d_out, int out_size, void* d_ws, size_t ws_size,
                              hipStream_t stream) {
    (void)in_sizes; (void)n_in; (void)out_size; (void)ws_size;
    const float* feat  = (const float*)d_in[0];
    const float* label = (const float*)d_in[1];
    const float* wgt   = (const float*)d_in[2];
    const float* bias  = (const float*)d_in[3];
    float* out = (float*)d_out;           // [moved 128^3 | flow 3*128^3]
    float* pos_flow = out + DHW;          // reuse flow region as conv scratch
    float* fA = (float*)d_ws;             // 3 * 64^3
    float* fB = fA + 3 * N3;

    // 1. conv3d via WMMA -> pos_flow
    dim3 cgrid(DSZ / TX, DSZ / TY, DSZ / TZ);
    conv3d_wmma_kernel<<<cgrid, 256, 0, stream>>>(feat, wgt, bias, pos_flow);

    // 2. downsample + prescale -> fA
    downsample_kernel<<<(N3 + 255) / 256, 256, 0, stream>>>(pos_flow, fA);

    // 3. seven scaling-and-squaring steps (ping-pong), ends in fB
    float* bufs[2] = {fA, fB};
    for (int i = 0; i < 7; ++i)
        vecint_step_kernel<<<(N3 + 255) / 256, 256, 0, stream>>>(bufs[i & 1], bufs[(i + 1) & 1]);

    // 4. upsample flow + warp label -> d_out (overwrites pos_flow region with final flow)
    finalize_kernel<<<(DHW + 255) / 256, 256, 0, stream>>>(fB, label, out);
}